// Deformer_50173807952232
// MI455X (gfx1250) — compile-verified
//
#include <hip/hip_runtime.h>

typedef __attribute__((ext_vector_type(2))) float v2f;
typedef __attribute__((ext_vector_type(8))) float v8f;

#define NB 256
#define NJ 24
#define NV 6890

// ---------------------------------------------------------------------------
// Kernel 1: forward kinematics + rel_transforms + posed_joints.
// One thread per batch (tiny work; correctness only).
// ---------------------------------------------------------------------------
__global__ void fk_kernel(const float* __restrict__ rot,
                          const float* __restrict__ jts,
                          float* __restrict__ posed_joints,
                          float* __restrict__ rel_tf) {
    constexpr int P[NJ] = {0,0,0,0,1,2,3,4,5,6,7,8,9,9,9,12,13,14,16,17,18,19,20,21};
    const int b = blockIdx.x * blockDim.x + threadIdx.x;
    if (b >= NB) return;

    float Jt[NJ][3];
#pragma unroll
    for (int j = 0; j < NJ; ++j) {
        Jt[j][0] = jts[b * NJ * 3 + j * 3 + 0];
        Jt[j][1] = jts[b * NJ * 3 + j * 3 + 1];
        Jt[j][2] = jts[b * NJ * 3 + j * 3 + 2];
    }

    // A[j] = top 3 rows of the 4x4 chain transform: [R | t], row-major 3x4.
    float A[NJ][12];
    const float* R0 = rot + (size_t)b * NJ * 9;
#pragma unroll
    for (int r = 0; r < 3; ++r) {
        A[0][r * 4 + 0] = R0[r * 3 + 0];
        A[0][r * 4 + 1] = R0[r * 3 + 1];
        A[0][r * 4 + 2] = R0[r * 3 + 2];
        A[0][r * 4 + 3] = Jt[0][r];
    }
#pragma unroll
    for (int j = 1; j < NJ; ++j) {
        const int p = P[j];
        const float rl0 = Jt[j][0] - Jt[p][0];
        const float rl1 = Jt[j][1] - Jt[p][1];
        const float rl2 = Jt[j][2] - Jt[p][2];
        const float* Rj = rot + ((size_t)b * NJ + j) * 9;
#pragma unroll
        for (int r = 0; r < 3; ++r) {
            const float a0 = A[p][r * 4 + 0];
            const float a1 = A[p][r * 4 + 1];
            const float a2 = A[p][r * 4 + 2];
            const float a3 = A[p][r * 4 + 3];
            A[j][r * 4 + 0] = a0 * Rj[0] + a1 * Rj[3] + a2 * Rj[6];
            A[j][r * 4 + 1] = a0 * Rj[1] + a1 * Rj[4] + a2 * Rj[7];
            A[j][r * 4 + 2] = a0 * Rj[2] + a1 * Rj[5] + a2 * Rj[8];
            A[j][r * 4 + 3] = a0 * rl0  + a1 * rl1  + a2 * rl2 + a3;
        }
    }

#pragma unroll
    for (int j = 0; j < NJ; ++j) {
        const float jx = Jt[j][0], jy = Jt[j][1], jz = Jt[j][2];
        float* out = rel_tf + (size_t)b * NJ * 16 + j * 16;
#pragma unroll
        for (int r = 0; r < 3; ++r) {
            posed_joints[b * NJ * 3 + j * 3 + r] = A[j][r * 4 + 3];
            const float tj = A[j][r * 4 + 0] * jx + A[j][r * 4 + 1] * jy +
                             A[j][r * 4 + 2] * jz;
            out[r * 4 + 0] = A[j][r * 4 + 0];
            out[r * 4 + 1] = A[j][r * 4 + 1];
            out[r * 4 + 2] = A[j][r * 4 + 2];
            out[r * 4 + 3] = A[j][r * 4 + 3] - tj;
        }
        out[12] = 0.0f; out[13] = 0.0f; out[14] = 0.0f; out[15] = 1.0f;
    }
}

// ---------------------------------------------------------------------------
// Kernel 2: vertex skinning via V_WMMA_F32_16X16X4_F32.
// Per batch: (V x 24) @ (24 x 16) GEMM, K split into 6 steps of 4.
// Block = 256 threads = 8 waves; each wave = 32 vertices = 2 M-tiles.
// ---------------------------------------------------------------------------
__global__ void skin_kernel(const float* __restrict__ W,
                            const float* __restrict__ verts,
                            const float* __restrict__ rel_tf,
                            float* __restrict__ posed) {
    __shared__ float sR[NJ * 16];          // rel transforms for this batch (24x16)
    __shared__ float sT[8][2][16][16];     // per-wave blended transform tiles

    const int b    = blockIdx.y;
    const int tid  = threadIdx.x;
    const int wave = tid >> 5;
    const int lane = tid & 31;
    const int half = lane >> 4;            // 0: K0/K1, 1: K2/K3 (A & B fragments)
    const int ln   = lane & 15;
    const int vwave = blockIdx.x * 256 + wave * 32;

    // Stage this batch's 24x16 transform block into LDS.
    for (int i = tid; i < NJ * 16; i += 256) sR[i] = rel_tf[(size_t)b * NJ * 16 + i];
    __syncthreads();

    // Clamp OOB vertex rows (keeps EXEC all-1s for WMMA; rows discarded later).
    const int v0 = vwave + ln;
    const int v1 = vwave + 16 + ln;
    const int v0c = v0 < NV ? v0 : NV - 1;
    const int v1c = v1 < NV ? v1 : NV - 1;

    v8f acc0 = {};
    v8f acc1 = {};
#pragma unroll
    for (int kk = 0; kk < 6; ++kk) {
        const int j0 = kk * 4 + half * 2;
        // B fragment: 4x16 slice of rel transforms.
        v2f bf;
        bf.x = sR[j0 * 16 + ln];
        bf.y = sR[(j0 + 1) * 16 + ln];
        // A fragments: 16x4 weight tiles (8-byte aligned float2 loads).
        const v2f a0 = *reinterpret_cast<const v2f*>(W + (size_t)v0c * NJ + j0);
        const v2f a1 = *reinterpret_cast<const v2f*>(W + (size_t)v1c * NJ + j0);
        acc0 = __builtin_amdgcn_wmma_f32_16x16x4_f32(false, a0, false, bf,
                                                     (short)0, acc0, false, false);
        acc1 = __builtin_amdgcn_wmma_f32_16x16x4_f32(false, a1, false, bf,
                                                     (short)0, acc1, false, false);
    }

    // Spill D tiles to LDS (C/D layout: VGPR r -> M = r + 8*half, N = lane&15).
#pragma unroll
    for (int r = 0; r < 8; ++r) {
        sT[wave][0][r + half * 8][ln] = acc0[r];
        sT[wave][1][r + half * 8][ln] = acc1[r];
    }
    __syncthreads();

    // Epilogue: one lane per vertex applies its blended 3x4 transform.
    const int v = vwave + lane;
    if (v < NV) {
        const float* T = &sT[wave][lane >> 4][lane & 15][0];
        const size_t base = ((size_t)b * NV + v) * 3;
        const float vx = verts[base + 0];
        const float vy = verts[base + 1];
        const float vz = verts[base + 2];
#pragma unroll
        for (int m = 0; m < 3; ++m) {
            posed[base + m] = T[4 * m + 0] * vx + T[4 * m + 1] * vy +
                              T[4 * m + 2] * vz + T[4 * m + 3];
        }
    }
}

extern "C" void kernel_launch(void* const* d_in, const int* in_sizes, int n_in,
                              void* d_out, int out_size, void* d_ws, size_t ws_size,
                              hipStream_t stream) {
    const float* rot   = (const float*)d_in[0];   // (B,J,3,3)
    const float* jts   = (const float*)d_in[1];   // (B,J,3)
    const float* verts = (const float*)d_in[2];   // (B,V,3)
    const float* W     = (const float*)d_in[3];   // (V,J)

    float* out          = (float*)d_out;
    float* posed_verts  = out;                                   // B*V*3
    float* posed_joints = out + (size_t)NB * NV * 3;             // B*J*3
    float* rel_tf       = posed_joints + (size_t)NB * NJ * 3;    // B*J*16

    fk_kernel<<<dim3(4), dim3(64), 0, stream>>>(rot, jts, posed_joints, rel_tf);

    dim3 grid((NV + 255) / 256, NB);
    skin_kernel<<<grid, dim3(256), 0, stream>>>(W, verts, rel_tf, posed_verts);
}